// GraphConvolutionLayer_14181982011963
// MI455X (gfx1250) — compile-verified
//
#include <hip/hip_runtime.h>
#include <hip/hip_bf16.h>

typedef __attribute__((ext_vector_type(2))) float v2f;
typedef __attribute__((ext_vector_type(8))) float v8f;

#define D_IN  256
#define D_OUT 256

#if defined(__has_builtin)
#if __has_builtin(__builtin_amdgcn_wmma_f32_16x16x4_f32)
#define HAVE_WMMA_F32X4 1
#endif
#endif

// ---------------------------------------------------------------------------
// Kernel 1: dense GEMM  xw = x @ W   using V_WMMA_F32_16X16X4_F32 (exact f32).
// One wave (32 lanes) computes a 16(M) x 64(N) tile: 4 f32 accumulators,
// looping K in steps of 4.  Grid = (N_NODES/16, D_OUT/64), block = 32.
// No divergence anywhere -> EXEC is all ones as WMMA requires.
//
// ISA VGPR layouts (CDNA5 ISA §7.12.2):
//   A (16x4 f32):  lane l -> row M = l&15 ;  VGPR0/1 hold K = 2*(l>>4)+{0,1}
//   B (4x16 f32):  lane l -> col N = l&15 ;  VGPR0/1 hold K = 2*(l>>4)+{0,1}
//   C/D (16x16 f32): VGPR v, lane l -> M = v + 8*(l>>4), N = l&15
// ---------------------------------------------------------------------------
__global__ void __launch_bounds__(32)
gcn_gemm_wmma_kernel(const float* __restrict__ x,
                     const float* __restrict__ w,
                     float* __restrict__ xw)
{
    const int lane = threadIdx.x & 31;
    const int m0   = blockIdx.x * 16;   // row tile base
    const int n0   = blockIdx.y * 64;   // col tile base

    const int mrow  = m0 + (lane & 15);        // A-frag row for this lane
    const int koff  = (lane >> 4) * 2;         // A/B-frag K sub-offset (0 or 2)
    const int ncol  = n0 + (lane & 15);        // B-frag column for this lane

#if HAVE_WMMA_F32X4
    v8f acc0 = {}, acc1 = {}, acc2 = {}, acc3 = {};

    const float* __restrict__ arow = x + (size_t)mrow * D_IN;

    for (int k0 = 0; k0 < D_IN; k0 += 4) {
        // A fragment: x[mrow][k0+koff], x[mrow][k0+koff+1]  (contiguous -> b64)
        v2f a;
        a.x = arow[k0 + koff + 0];
        a.y = arow[k0 + koff + 1];

        // B fragments for the 4 column sub-tiles: W[k][ncol + 16*t]
        const float* __restrict__ brow0 = w + (size_t)(k0 + koff + 0) * D_OUT + ncol;
        const float* __restrict__ brow1 = w + (size_t)(k0 + koff + 1) * D_OUT + ncol;
        v2f b0, b1, b2, b3;
        b0.x = brow0[ 0]; b0.y = brow1[ 0];
        b1.x = brow0[16]; b1.y = brow1[16];
        b2.x = brow0[32]; b2.y = brow1[32];
        b3.x = brow0[48]; b3.y = brow1[48];

        acc0 = __builtin_amdgcn_wmma_f32_16x16x4_f32(false, a, false, b0,
                                                     (short)0, acc0, false, false);
        acc1 = __builtin_amdgcn_wmma_f32_16x16x4_f32(false, a, false, b1,
                                                     (short)0, acc1, false, false);
        acc2 = __builtin_amdgcn_wmma_f32_16x16x4_f32(false, a, false, b2,
                                                     (short)0, acc2, false, false);
        acc3 = __builtin_amdgcn_wmma_f32_16x16x4_f32(false, a, false, b3,
                                                     (short)0, acc3, false, false);
    }

    // D layout: VGPR v, lane l -> row M = v + 8*(l>>4), col N = l&15
    const int dbase_m = m0 + 8 * (lane >> 4);
    const int dbase_n = n0 + (lane & 15);
#pragma unroll
    for (int v = 0; v < 8; ++v) {
        float* __restrict__ drow = xw + (size_t)(dbase_m + v) * D_OUT + dbase_n;
        drow[ 0] = acc0[v];
        drow[16] = acc1[v];
        drow[32] = acc2[v];
        drow[48] = acc3[v];
    }
#else
    // Scalar fallback (keeps the build green if the builtin is absent).
    for (int mm = 0; mm < 16; ++mm) {
        for (int t = 0; t < 2; ++t) {
            const int n = n0 + lane + 32 * t;
            float s = 0.f;
            for (int k = 0; k < D_IN; ++k)
                s = fmaf(x[(size_t)(m0 + mm) * D_IN + k],
                         w[(size_t)k * D_OUT + n], s);
            xw[(size_t)(m0 + mm) * D_OUT + n] = s;
        }
    }
    (void)mrow; (void)koff; (void)ncol;
#endif
}

// ---------------------------------------------------------------------------
// Kernel 2: COO SpMM with atomic scatter.
//   out[row[e]][:] += vals[e] * xw[col[e]][:]
// 64 lanes per edge, float4 per lane -> each edge's 1KB row gather is fully
// coalesced.  out (51.2 MB) fits in the 192 MB L2, so the f32 atomics resolve
// in L2 rather than HBM.
// ---------------------------------------------------------------------------
__global__ void __launch_bounds__(256)
gcn_spmm_atomic_kernel(const int*   __restrict__ erow,
                       const int*   __restrict__ ecol,
                       const float* __restrict__ evals,
                       const float* __restrict__ xw,
                       float* __restrict__ out,
                       int n_edges)
{
    const int gid = blockIdx.x * blockDim.x + threadIdx.x;
    const int e   = gid >> 6;           // 64 threads per edge
    if (e >= n_edges) return;
    const int f   = (gid & 63) << 2;    // 4 features per thread

    const int   r = erow[e];
    const int   c = ecol[e];
    const float v = evals[e];

    const float4 src = *(const float4*)(xw + (size_t)c * D_OUT + f);
    float* __restrict__ dst = out + (size_t)r * D_OUT + f;

    atomicAdd(dst + 0, v * src.x);
    atomicAdd(dst + 1, v * src.y);
    atomicAdd(dst + 2, v * src.z);
    atomicAdd(dst + 3, v * src.w);
}

// ---------------------------------------------------------------------------
// Kernel 3: epilogue  out = relu(out + bias)  (in place, float4 vectorized)
// ---------------------------------------------------------------------------
__global__ void __launch_bounds__(256)
gcn_bias_relu_kernel(float* __restrict__ out,
                     const float* __restrict__ bias,
                     int n_nodes)
{
    const int gid  = blockIdx.x * blockDim.x + threadIdx.x;
    const int node = gid >> 6;          // 64 threads per node row
    if (node >= n_nodes) return;
    const int f    = (gid & 63) << 2;

    float4 o = *(float4*)(out + (size_t)node * D_OUT + f);
    const float4 b = *(const float4*)(bias + f);
    o.x = fmaxf(o.x + b.x, 0.f);
    o.y = fmaxf(o.y + b.y, 0.f);
    o.z = fmaxf(o.z + b.z, 0.f);
    o.w = fmaxf(o.w + b.w, 0.f);
    *(float4*)(out + (size_t)node * D_OUT + f) = o;
}

// ---------------------------------------------------------------------------
// Host-side launch.  Inputs (setup_inputs order):
//   d_in[0] x          [N_NODES, 256] f32
//   d_in[1] edge_row   [E] i32
//   d_in[2] edge_col   [E] i32
//   d_in[3] edge_vals  [E] f32
//   d_in[4] weight     [256, 256] f32
//   d_in[5] bias       [256] f32
// d_out: [N_NODES, 256] f32.  d_ws: scratch for xw (N_NODES*256 f32).
// ---------------------------------------------------------------------------
extern "C" void kernel_launch(void* const* d_in, const int* in_sizes, int n_in,
                              void* d_out, int out_size, void* d_ws, size_t ws_size,
                              hipStream_t stream)
{
    const float* x     = (const float*)d_in[0];
    const int*   erow  = (const int*)  d_in[1];
    const int*   ecol  = (const int*)  d_in[2];
    const float* evals = (const float*)d_in[3];
    const float* w     = (const float*)d_in[4];
    const float* bias  = (const float*)d_in[5];
    float*       out   = (float*)d_out;
    float*       xw    = (float*)d_ws;

    const int n_nodes = in_sizes[0] / D_IN;
    const int n_edges = in_sizes[1];

    // 1) zero the accumulation target (capturable as a memset node)
    hipMemsetAsync(d_out, 0, (size_t)out_size * sizeof(float), stream);

    // 2) WMMA GEMM: xw = x @ W   (one wave per 16x64 tile)
    {
        dim3 grid(n_nodes / 16, D_OUT / 64);
        gcn_gemm_wmma_kernel<<<grid, 32, 0, stream>>>(x, w, xw);
    }

    // 3) SpMM: atomic scatter over edges (64 threads/edge)
    {
        const long long total  = (long long)n_edges * 64;
        const int       blocks = (int)((total + 255) / 256);
        gcn_spmm_atomic_kernel<<<blocks, 256, 0, stream>>>(erow, ecol, evals,
                                                           xw, out, n_edges);
    }

    // 4) bias + ReLU epilogue (in place on d_out)
    {
        const long long total  = (long long)n_nodes * 64;
        const int       blocks = (int)((total + 255) / 256);
        gcn_bias_relu_kernel<<<blocks, 256, 0, stream>>>(out, bias, n_nodes);
    }
}